// TeacherRetrieverPool_9526237462634
// MI455X (gfx1250) — compile-verified
//
#include <hip/hip_runtime.h>
#include <stdint.h>

// Problem constants (match reference setup_inputs)
constexpr int   B      = 60;
constexpr int   T      = 4;
constexpr int   K      = 100;
constexpr int   TK     = T * K;        // 400 entries per query
constexpr int   NDOCS  = 1000000;
constexpr float RRFK   = 60.0f;
constexpr int   SORT_N = 512;          // bitonic width (>= TK)
constexpr int   HASH_N = 1024;         // LDS hash table slots

// Vector/addrspace types for the gfx1250 async global->LDS builtin:
// first param is 'int4 __device__*' (addrspace(1)), second is LDS (addrspace(3)).
typedef __attribute__((ext_vector_type(4))) int v4i;
typedef __attribute__((address_space(1))) v4i gv4i_t;
typedef __attribute__((address_space(3))) v4i lv4i_t;

// ---------------------------------------------------------------------------
// Kernel 1: one block per query.
//  - LDS hash table accumulates duplicate doc ids (ds atomics).
//  - Pack (score, id) into u64 key = float_bits(score)<<32 | ~id, so a plain
//    descending u64 sort == (score desc, id asc)  [scores are > 0].
//  - Bitonic sort 512 keys; write the u scored ids to the front of the
//    output row; bitonic-sort the ids ascending and stash (padded with
//    INT_MAX) + count into workspace for the fill kernel.
// ---------------------------------------------------------------------------
__global__ __launch_bounds__(512)
void rrf_build_kernel(const int* __restrict__ idx,
                      const float* __restrict__ rnk,
                      const float* __restrict__ wgt,
                      int* __restrict__ out,       // [B, NDOCS]
                      int* __restrict__ ws_ids,    // [B, SORT_N]
                      int* __restrict__ ws_cnt)    // [B]
{
  __shared__ int                hid[HASH_N];
  __shared__ float              hsc[HASH_N];
  __shared__ unsigned long long keys[SORT_N];
  __shared__ int                sids[SORT_N];
  __shared__ int                cnt;

  const int b   = blockIdx.x;
  const int tid = threadIdx.x;

  for (int s = tid; s < HASH_N; s += 512) { hid[s] = -1; hsc[s] = 0.0f; }
  if (tid == 0) cnt = 0;
  __syncthreads();

  if (tid < TK) {
    const int   id = idx[b * TK + tid];
    const int   t  = tid / K;
    const float s  = wgt[t] / (RRFK + rnk[b * TK + tid]);
    unsigned h = (((unsigned)id * 2654435761u) >> 16) & (HASH_N - 1);
    for (;;) {
      int prev = atomicCAS(&hid[h], -1, id);
      if (prev == -1 || prev == id) { atomicAdd(&hsc[h], s); break; }
      h = (h + 1) & (HASH_N - 1);
    }
  }
  __syncthreads();

  // Compact occupied slots into packed sort keys.
  for (int s = tid; s < HASH_N; s += 512) {
    const int id = hid[s];
    if (id != -1) {
      const int p = atomicAdd(&cnt, 1);
      const unsigned sb = __float_as_uint(hsc[s]);   // positive float: uint-monotone
      keys[p] = ((unsigned long long)sb << 32) | (unsigned)(~(unsigned)id);
    }
  }
  __syncthreads();
  const int u = cnt;                                  // unique scored docs (<= 400)
  for (int p = u + tid; p < SORT_N; p += 512) keys[p] = 0ull;  // sentinel: sorts last
  __syncthreads();

  // Bitonic sort, descending.
  for (int k = 2; k <= SORT_N; k <<= 1)
    for (int j = k >> 1; j > 0; j >>= 1) {
      const int i = tid, ixj = i ^ j;
      if (ixj > i) {
        const unsigned long long a = keys[i], c = keys[ixj];
        const bool desc = ((i & k) == 0);
        if (desc ? (a < c) : (a > c)) { keys[i] = c; keys[ixj] = a; }
      }
      __syncthreads();
    }

  // Front of the output row = scored docs in final order; build id table.
  {
    const int id = (tid < u) ? (int)(~(unsigned)keys[tid]) : 0x7FFFFFFF;
    sids[tid] = id;
    if (tid < u) __builtin_nontemporal_store(id, &out[(size_t)b * NDOCS + tid]);
  }
  __syncthreads();

  // Bitonic sort ids ascending (pads INT_MAX stay at the end).
  for (int k = 2; k <= SORT_N; k <<= 1)
    for (int j = k >> 1; j > 0; j >>= 1) {
      const int i = tid, ixj = i ^ j;
      if (ixj > i) {
        const int a = sids[i], c = sids[ixj];
        const bool asc = ((i & k) == 0);
        if (asc ? (a > c) : (a < c)) { sids[i] = c; sids[ixj] = a; }
      }
      __syncthreads();
    }

  ws_ids[b * SORT_N + tid] = sids[tid];
  if (tid == 0) ws_cnt[b] = u;
}

// ---------------------------------------------------------------------------
// Kernel 2: streaming fill of the argsort tail (the HBM-bound part).
// Output position of unscored doc d is  u + d - |{scored ids < d}|.
// The 512-entry sorted id table is staged into LDS with the gfx1250
// async global->LDS path; each lane does a 9-step branchless LDS binary
// search, then a nontemporal b32 store (addresses are monotone per lane
// group => coalesced).
// ---------------------------------------------------------------------------
constexpr int FILL_T = 256;
constexpr int FILL_I = 16;
constexpr int CHUNK  = FILL_T * FILL_I;   // 4096 docs per block

__global__ __launch_bounds__(FILL_T)
void rrf_fill_kernel(const int* __restrict__ ws_ids,
                     const int* __restrict__ ws_cnt,
                     int* __restrict__ out)
{
  __shared__ int sids[SORT_N];
  __shared__ int su;
  const int b    = blockIdx.y;
  const int tid  = threadIdx.x;
  const int base = blockIdx.x * CHUNK;

#if __has_builtin(__builtin_amdgcn_global_load_async_to_lds_b128)
  if (tid < SORT_N / 4) {   // 128 lanes x 16B = 2KB table
    const int* g = ws_ids + b * SORT_N + tid * 4;
    __builtin_amdgcn_global_load_async_to_lds_b128(
        (gv4i_t*)(g),
        (lv4i_t*)(&sids[tid * 4]),
        0, 0);
  }
#if __has_builtin(__builtin_amdgcn_s_wait_asynccnt)
  __builtin_amdgcn_s_wait_asynccnt(0);
#else
  asm volatile("s_wait_asynccnt 0" ::: "memory");
#endif
#else
  for (int i = tid; i < SORT_N; i += FILL_T)
    sids[i] = ws_ids[b * SORT_N + i];
#endif
  if (tid == 0) su = ws_cnt[b];
  __syncthreads();
  const int u = su;

  int* orow = out + (size_t)b * NDOCS;
#pragma unroll
  for (int it = 0; it < FILL_I; ++it) {
    const int d = base + it * FILL_T + tid;
    if (d < NDOCS) {
      int pos = 0;                       // branchless lower_bound over 512 slots
#pragma unroll
      for (int s = SORT_N / 2; s > 0; s >>= 1)
        if (sids[pos + s - 1] < d) pos += s;
      if (sids[pos] != d)                // skip scored docs (already at front)
        __builtin_nontemporal_store(d, &orow[u + d - pos]);
    }
  }
}

// ---------------------------------------------------------------------------
// Kernel 3: positive_idx[b] = order[b][pos_positions[b]]  (pos < 5, already
// written by kernels 1/2; same stream => ordered).
// ---------------------------------------------------------------------------
__global__ void rrf_pos_kernel(const int* __restrict__ posp,
                               const int* __restrict__ order,
                               int* __restrict__ out_pos)
{
  const int b = threadIdx.x;
  if (b < B) {
    const int p = posp[b];
    out_pos[b] = order[(size_t)b * NDOCS + p];
  }
}

extern "C" void kernel_launch(void* const* d_in, const int* in_sizes, int n_in,
                              void* d_out, int out_size, void* d_ws, size_t ws_size,
                              hipStream_t stream) {
  const int*   idx  = (const int*)d_in[0];   // [B,T,K] int32
  const float* rnk  = (const float*)d_in[1]; // [B,T,K] f32
  const float* wgt  = (const float*)d_in[2]; // [T] f32
  const int*   posp = (const int*)d_in[3];   // [B] int32
  // d_in[4] = num_docs scalar (compile-time constant here)

  int* out    = (int*)d_out;                 // order [B*NDOCS] ++ positive_idx [B]
  int* ws_ids = (int*)d_ws;                  // [B, SORT_N]
  int* ws_cnt = ws_ids + B * SORT_N;         // [B]   (~123 KB total workspace)

  rrf_build_kernel<<<B, 512, 0, stream>>>(idx, rnk, wgt, out, ws_ids, ws_cnt);

  dim3 g((NDOCS + CHUNK - 1) / CHUNK, B);
  rrf_fill_kernel<<<g, FILL_T, 0, stream>>>(ws_ids, ws_cnt, out);

  rrf_pos_kernel<<<1, 64, 0, stream>>>(posp, out, out + (size_t)B * NDOCS);
}